// BayesianGFL_62895501082688
// MI455X (gfx1250) — compile-verified
//
#include <hip/hip_runtime.h>

// BayesianGFL graph filter:
//   xs = x.sum(1); U[r,i] += vals^fact(i) * xs[cols]; out = U @ coeff.T
// Exponents are [1,1,2,6] -> taps 0 and 1 are identical, store only 3 taps.

typedef __attribute__((ext_vector_type(2))) float v2f;
typedef __attribute__((ext_vector_type(4))) float v4f;
typedef __attribute__((ext_vector_type(8))) float v8f;

#define F_IN  64
#define F_OUT 64

// ---------------------------------------------------------------------------
// Kernel 1: xs[n] = sum_f x[n,f]; also zero the accumulator U[n, 0..3].
// ---------------------------------------------------------------------------
__global__ __launch_bounds__(256) void gfl_rowsum_zero(
    const float* __restrict__ x, float* __restrict__ xs,
    float* __restrict__ U, int n) {
  int i = blockIdx.x * blockDim.x + threadIdx.x;
  if (i >= n) return;
  const float4* xr = (const float4*)(x + (size_t)i * F_IN);
  float s = 0.f;
#pragma unroll
  for (int j = 0; j < F_IN / 4; ++j) {
    float4 v = xr[j];
    s += (v.x + v.y) + (v.z + v.w);
  }
  xs[i] = s;
  float4 z = make_float4(0.f, 0.f, 0.f, 0.f);
  *(float4*)(U + (size_t)i * 4) = z;
}

// ---------------------------------------------------------------------------
// Kernel 2: per-edge scatter, 4 edges per thread with 128-bit loads.
//   g = xs[col];  U[row,0] += v*g;  U[row,1] += v^2*g;  U[row,2] += v^6*g
// 3 no-return fp32 atomics/edge into a 1.6MB L2-resident table.
// ---------------------------------------------------------------------------
__device__ __forceinline__ void edge_accum(int r, int c, float v,
                                           const float* __restrict__ xs,
                                           float* __restrict__ U) {
  float g  = xs[c];
  float v2 = v * v;
  float v6 = v2 * v2 * v2;
  float* u = U + (size_t)r * 4;
  unsafeAtomicAdd(u + 0, v  * g);
  unsafeAtomicAdd(u + 1, v2 * g);
  unsafeAtomicAdd(u + 2, v6 * g);
}

__global__ __launch_bounds__(256) void gfl_edge_scatter(
    const int* __restrict__ rows, const int* __restrict__ cols,
    const float* __restrict__ vals, const float* __restrict__ xs,
    float* __restrict__ U, int e_total) {
  int t  = blockIdx.x * blockDim.x + threadIdx.x;
  int e0 = t * 4;
  if (e0 + 3 < e_total) {                 // fast path: 16B coalesced loads
    int4   r4 = *(const int4*)(rows + e0);
    int4   c4 = *(const int4*)(cols + e0);
    float4 v4 = *(const float4*)(vals + e0);
    edge_accum(r4.x, c4.x, v4.x, xs, U);
    edge_accum(r4.y, c4.y, v4.y, xs, U);
    edge_accum(r4.z, c4.z, v4.z, xs, U);
    edge_accum(r4.w, c4.w, v4.w, xs, U);
  } else {                                // ragged tail (empty when E % 4 == 0)
    for (int e = e0; e < e_total; ++e)
      edge_accum(rows[e], cols[e], vals[e], xs, U);
  }
}

// ---------------------------------------------------------------------------
// Kernel 3: out[16-row tile, 0..63] = A(16x4) @ B(4x16) x 4 column blocks
// via V_WMMA_F32_16X16X4_F32.  One wave per 16-node tile.
//
// A layout (ISA 7.12.2, f32 16x4): lanes 0-15 -> {K0,K1}, lanes 16-31 -> {K2,K3}.
// K-dim values are [t1, t1, t2, t6] (duplicate tap reconstructed here), with
// stored U = [t1, t2, t6, pad].
// ---------------------------------------------------------------------------
__global__ __launch_bounds__(128) void gfl_wmma_project(
    const float* __restrict__ U, const float* __restrict__ coeff,
    float* __restrict__ out, int n, int ntiles) {
  int wave = (int)((blockIdx.x * blockDim.x + threadIdx.x) >> 5);  // wave-uniform
  if (wave >= ntiles) return;                                      // uniform exit
  int lane = (int)(threadIdx.x & 31);
  int half = lane >> 4;   // 0: K=0,1   1: K=2,3
  int l16  = lane & 15;

  // ---- A operand: U tile row (clamped load for a possible ragged tail) ----
  int row    = wave * 16 + l16;
  int rclamp = row < n ? row : 0;
  v4f u4 = *(const v4f*)(U + (size_t)rclamp * 4);   // {t1, t2, t6, pad}
  v2f a;
  a.x = half ? u4.y : u4.x;   // K0 -> t1 | K2 -> t2
  a.y = half ? u4.z : u4.x;   // K1 -> t1 | K3 -> t6
  int  k0        = half * 2;
  int  base_m    = wave * 16 + half * 8;
  bool full_tile = (wave * 16 + 16) <= n;           // wave-uniform

#pragma unroll
  for (int cb = 0; cb < 4; ++cb) {
    // ---- B operand: coeff.T column block; B[k][ncol] = coeff[ncol][k] ----
    int ncol = cb * 16 + l16;
    v2f b = *(const v2f*)(coeff + (size_t)ncol * 4 + k0);  // 8B-aligned (k0 in {0,2})
    v8f c = {};
    c = __builtin_amdgcn_wmma_f32_16x16x4_f32(
        /*neg_a=*/false, a, /*neg_b=*/false, b,
        /*c_mod=*/(short)0, c, /*reuse_a=*/false, /*reuse_b=*/false);
    // ---- D layout: lane half h, VGPR j -> row base_m+j, col cb*16+l16 ----
    int col = cb * 16 + l16;
    if (full_tile) {
      // single base address, 8 stores with immediate offsets (j*256B)
      float* p = out + (size_t)base_m * F_OUT + col;
#pragma unroll
      for (int j = 0; j < 8; ++j) p[j * F_OUT] = c[j];
    } else {
#pragma unroll
      for (int j = 0; j < 8; ++j) {
        int m = base_m + j;
        if (m < n) out[(size_t)m * F_OUT + col] = c[j];
      }
    }
  }
}

// ---------------------------------------------------------------------------
extern "C" void kernel_launch(void* const* d_in, const int* in_sizes, int n_in,
                              void* d_out, int out_size, void* d_ws, size_t ws_size,
                              hipStream_t stream) {
  const float* x     = (const float*)d_in[0];
  const int*   rows  = (const int*)d_in[1];
  const int*   cols  = (const int*)d_in[2];
  const float* vals  = (const float*)d_in[3];
  const float* coeff = (const float*)d_in[4];
  // d_in[5] = ind, unused by the math.

  int N = in_sizes[0] / F_IN;   // 100000
  int E = in_sizes[1];          // 6400000

  float* xs = (float*)d_ws;     // N floats
  float* U  = xs + N;           // N*4 floats (tap accumulators, slot 3 = pad)
  float* out = (float*)d_out;

  gfl_rowsum_zero<<<(N + 255) / 256, 256, 0, stream>>>(x, xs, U, N);

  int nquadthreads = (E + 3) / 4;
  gfl_edge_scatter<<<(nquadthreads + 255) / 256, 256, 0, stream>>>(
      rows, cols, vals, xs, U, E);

  int ntiles = (N + 15) / 16;   // 6250
  int waves_per_block = 4;      // 128 threads
  int blocks = (ntiles + waves_per_block - 1) / waves_per_block;
  gfl_wmma_project<<<blocks, waves_per_block * 32, 0, stream>>>(U, coeff, out, N, ntiles);
}